// Critic_78228534329751
// MI455X (gfx1250) — compile-verified
//
#include <hip/hip_runtime.h>
#include <hip/hip_bf16.h>

// ---------------------------------------------------------------------------
// GCN (4 layers) for MI455X / gfx1250, wave32.
// Dense transforms use V_WMMA_F32_16X16X4_F32 (native f32 matrix op, keeps
// reference numerics). Edge scatter/gather stays in the 192MB L2 (feature
// table is 6.4MB), so HBM traffic is just the edge lists + norm (~0.3GB).
// ---------------------------------------------------------------------------

typedef __attribute__((ext_vector_type(2))) float v2f;
typedef __attribute__((ext_vector_type(8))) float v8f;

#define GHID 16

// ---------------- norm precompute ----------------

__global__ void k_init_deg(float* __restrict__ deg, int n) {
  int i = blockIdx.x * blockDim.x + threadIdx.x;
  if (i < n) deg[i] = 1.0f;  // self-loop weight
}

__global__ void k_accum_deg(const int* __restrict__ dst, const float* __restrict__ w,
                            float* __restrict__ deg, long long e) {
  long long i = (long long)blockIdx.x * blockDim.x + threadIdx.x;
  if (i < e) atomicAdd(&deg[dst[i]], w[i]);
}

__global__ void k_deg_to_dis(float* __restrict__ deg, int n) {
  int i = blockIdx.x * blockDim.x + threadIdx.x;
  if (i < n) {
    float d = deg[i];
    deg[i] = (d > 0.0f) ? rsqrtf(d) : 0.0f;  // deg >= 1 always, guard mirrors ref
  }
}

__global__ void k_edge_norm(const int* __restrict__ src, const int* __restrict__ dst,
                            const float* __restrict__ w, const float* __restrict__ dis,
                            float* __restrict__ norm, long long e) {
  long long i = (long long)blockIdx.x * blockDim.x + threadIdx.x;
  if (i < e) norm[i] = dis[src[i]] * w[i] * dis[dst[i]];
}

// ---------------- dense transform: [n,K] @ [K,16] via WMMA f32 16x16x4 ------
// One wave per 16-row tile; EXEC all-ones inside the wave (WMMA requirement).
// A layout (16x4): lane<16 -> K={k,k+1}, lane>=16 -> K={k+2,k+3}, M = lane&15.
// B layout (4x16): same K split, N = lane&15.
// C/D: VGPR r -> row (lane<16 ? r : r+8), col = lane&15.

__global__ void k_gemm_wmma16(const float* __restrict__ X, const float* __restrict__ W,
                              float* __restrict__ Y, int n, int K) {
  int wave = (int)((blockIdx.x * (long long)blockDim.x + threadIdx.x) >> 5);
  int lane = threadIdx.x & 31;
  int rowBase = wave * 16;
  if (rowBase >= n) return;  // whole-wave branch; EXEC stays full

  const int m = lane & 15;         // row (A) / col (B,C)
  const int kh = (lane >> 4) * 2;  // K sub-offset: 0 or 2

  v8f c = {};
  const float* xrow = X + (size_t)(rowBase + m) * K;
  for (int k = 0; k < K; k += 4) {
    v2f a, b;
    a.x = xrow[k + kh];
    a.y = xrow[k + kh + 1];
    b.x = W[(k + kh) * GHID + m];
    b.y = W[(k + kh + 1) * GHID + m];
    c = __builtin_amdgcn_wmma_f32_16x16x4_f32(false, a, false, b, (short)0, c,
                                              false, false);
  }

  float* yout = Y + (size_t)rowBase * GHID;
  const int rOff = (lane >> 4) * 8;
#pragma unroll
  for (int r = 0; r < 8; ++r) yout[(size_t)(r + rOff) * GHID + m] = c[r];
}

// ---------------- aggregation (16-wide layers) ----------------

// out[i,j] = dis[i]^2 * y[i,j]   (self-loop message initializes accumulator)
__global__ void k_self_init16(const float* __restrict__ dis, const float* __restrict__ y,
                              float* __restrict__ out, int n16) {
  int g = blockIdx.x * blockDim.x + threadIdx.x;
  if (g < n16) {
    float d = dis[g >> 4];
    out[g] = d * d * y[g];
  }
}

// 16 lanes per edge: coalesced 64B gather of y[src], cacheline-local atomics.
__global__ void k_edge_agg16(const int* __restrict__ src, const int* __restrict__ dst,
                             const float* __restrict__ norm, const float* __restrict__ y,
                             float* __restrict__ out, long long e16) {
  long long g = (long long)blockIdx.x * blockDim.x + threadIdx.x;
  if (g >= e16) return;
  int e = (int)(g >> 4);
  int j = (int)(g & 15);
  float v = norm[e] * y[src[e] * GHID + j];
  atomicAdd(&out[dst[e] * GHID + j], v);
}

__global__ void k_bias_relu16(float* __restrict__ h, const float* __restrict__ b, int n16) {
  int g = blockIdx.x * blockDim.x + threadIdx.x;
  if (g < n16) h[g] = fmaxf(h[g] + b[g & 15], 0.0f);
}

// ---------------- layer 4 (out dim 1) ----------------

__global__ void k_gemv16(const float* __restrict__ h, const float* __restrict__ W4,
                         float* __restrict__ y, int n) {
  int i = blockIdx.x * blockDim.x + threadIdx.x;
  if (i >= n) return;
  float acc = 0.0f;
#pragma unroll
  for (int j = 0; j < GHID; ++j) acc += h[(size_t)i * GHID + j] * W4[j];
  y[i] = acc;
}

__global__ void k_self_init1(const float* __restrict__ dis, const float* __restrict__ y,
                             float* __restrict__ out, int n) {
  int i = blockIdx.x * blockDim.x + threadIdx.x;
  if (i < n) {
    float d = dis[i];
    out[i] = d * d * y[i];
  }
}

__global__ void k_edge_agg1(const int* __restrict__ src, const int* __restrict__ dst,
                            const float* __restrict__ norm, const float* __restrict__ y,
                            float* __restrict__ out, long long e) {
  long long i = (long long)blockIdx.x * blockDim.x + threadIdx.x;
  if (i < e) atomicAdd(&out[dst[i]], norm[i] * y[src[i]]);
}

__global__ void k_zero1(float* __restrict__ out) {
  if (blockIdx.x == 0 && threadIdx.x == 0) out[0] = 0.0f;
}

// bias + ReLU + scaled block reduction -> atomic into out[0]
__global__ void k_bias_relu_mean(const float* __restrict__ agg, const float* __restrict__ b4,
                                 float invN, float* __restrict__ out, int n) {
  __shared__ float sdata[256];
  int i = blockIdx.x * 256 + threadIdx.x;
  float v = 0.0f;
  if (i < n) v = fmaxf(agg[i] + b4[0], 0.0f) * invN;
  sdata[threadIdx.x] = v;
  __syncthreads();
  for (int s = 128; s > 0; s >>= 1) {
    if (threadIdx.x < s) sdata[threadIdx.x] += sdata[threadIdx.x + s];
    __syncthreads();
  }
  if (threadIdx.x == 0) atomicAdd(out, sdata[0]);
}

// ---------------------------------------------------------------------------

extern "C" void kernel_launch(void* const* d_in, const int* in_sizes, int n_in,
                              void* d_out, int out_size, void* d_ws, size_t ws_size,
                              hipStream_t stream) {
  const float* X  = (const float*)d_in[0];   // [N,128]
  const int*   eg = (const int*)d_in[1];     // [2,E]
  const float* w  = (const float*)d_in[2];   // [E]
  const float* W1 = (const float*)d_in[3];
  const float* b1 = (const float*)d_in[4];
  const float* W2 = (const float*)d_in[5];
  const float* b2 = (const float*)d_in[6];
  const float* W3 = (const float*)d_in[7];
  const float* b3 = (const float*)d_in[8];
  const float* W4 = (const float*)d_in[9];
  const float* b4 = (const float*)d_in[10];
  float* out = (float*)d_out;

  const int       N = in_sizes[0] / 128;
  const long long E = (long long)in_sizes[2];
  const int* src = eg;
  const int* dst = eg + E;

  // workspace carve-up (floats)
  float* deg  = (float*)d_ws;                  // N   (becomes dis after k_deg_to_dis)
  float* norm = deg + N;                       // E
  float* bufA = norm + E;                      // N*16 (GEMM output / messages)
  float* bufB = bufA + (size_t)N * GHID;       // N*16 (aggregation / layer input)

  const int B = 256;
  const int gN   = (N + B - 1) / B;
  const int gN16 = (N * GHID + B - 1) / B;
  const int gE   = (int)((E + B - 1) / B);
  const int gE16 = (int)((E * GHID + B - 1) / B);
  const int tiles  = (N + 15) / 16;            // one wave per tile, 8 waves/block
  const int gGemm  = (tiles + 7) / 8;

  // ---- norm precompute ----
  k_init_deg<<<gN, B, 0, stream>>>(deg, N);
  k_accum_deg<<<gE, B, 0, stream>>>(dst, w, deg, E);
  k_deg_to_dis<<<gN, B, 0, stream>>>(deg, N);
  k_edge_norm<<<gE, B, 0, stream>>>(src, dst, w, deg, norm, E);

  // ---- layer 1: [N,128] @ [128,16] ----
  k_gemm_wmma16<<<gGemm, 256, 0, stream>>>(X, W1, bufA, N, 128);
  k_self_init16<<<gN16, B, 0, stream>>>(deg, bufA, bufB, N * GHID);
  k_edge_agg16<<<gE16, B, 0, stream>>>(src, dst, norm, bufA, bufB, E * GHID);
  k_bias_relu16<<<gN16, B, 0, stream>>>(bufB, b1, N * GHID);

  // ---- layers 2,3: [N,16] @ [16,16] ----
  const float* Ws[2] = {W2, W3};
  const float* bs[2] = {b2, b3};
  for (int l = 0; l < 2; ++l) {
    k_gemm_wmma16<<<gGemm, 256, 0, stream>>>(bufB, Ws[l], bufA, N, GHID);
    k_self_init16<<<gN16, B, 0, stream>>>(deg, bufA, bufB, N * GHID);
    k_edge_agg16<<<gE16, B, 0, stream>>>(src, dst, norm, bufA, bufB, E * GHID);
    k_bias_relu16<<<gN16, B, 0, stream>>>(bufB, bs[l], N * GHID);
  }

  // ---- layer 4: [N,16] @ [16,1], aggregate, mean ----
  k_gemv16<<<gN, B, 0, stream>>>(bufB, W4, bufA, N);
  k_self_init1<<<gN, B, 0, stream>>>(deg, bufA, bufB, N);
  k_edge_agg1<<<gE, B, 0, stream>>>(src, dst, norm, bufA, bufB, E);
  k_zero1<<<1, 32, 0, stream>>>(out);
  k_bias_relu_mean<<<gN, 256, 0, stream>>>(bufB, b4, 1.0f / (float)N, out, N);
}